// StandardDP_60971355734709
// MI455X (gfx1250) — compile-verified
//
#include <hip/hip_runtime.h>
#include <stdint.h>

#define NEG_INF (-999999.0f)
#define BB 32
#define SS 512
#define KK 32

typedef float f32x4 __attribute__((ext_vector_type(4)));
typedef unsigned int u32x4 __attribute__((ext_vector_type(4)));
typedef int i32x4 __attribute__((ext_vector_type(4)));
typedef int i32x8 __attribute__((ext_vector_type(8)));

#if defined(__has_builtin)
#if __has_builtin(__builtin_amdgcn_tensor_load_to_lds) && __has_builtin(__builtin_amdgcn_s_wait_tensorcnt)
#define USE_TDM 1
#endif
#endif
#ifndef USE_TDM
#define USE_TDM 0
#endif

// ---------------------------------------------------------------------------
// Kernel 2 (emitted first so the disasm snippet shows the TDM path):
// sequential max-plus DP, one block per batch, 512 threads (16 wave32).
// Column chunks of label_table[b] staged by the Tensor Data Mover into LDS,
// double-buffered, pipelined with s_wait_tensorcnt. D# pad feature gives
// row stride 9 dwords -> the per-step column read is LDS bank-conflict-free.
// ---------------------------------------------------------------------------
#if USE_TDM
#define TILE_C 8        // columns (i) per TDM chunk
#define TILE_STRIDE 9   // 8 data dwords + 1 pad dword
#define TILE_DWORDS 4608
#define NCHUNK (SS / TILE_C)

__device__ __forceinline__ void tdm_issue(const float* gtile, unsigned lds_byte) {
  unsigned long long ga = (unsigned long long)(uintptr_t)gtile;
  u32x4 g0;
  g0.x = 1u;                                             // count=1, load, no gather
  g0.y = lds_byte;                                       // lds_addr (bytes)
  g0.z = (unsigned)(ga & 0xffffffffull);                 // global_addr[31:0]
  g0.w = (unsigned)((ga >> 32) & 0x01ffffffull)          // global_addr[56:32]
         | (2u << 30);                                   // type=2 (image)
  i32x8 g1;
  g1[0] = (2 << 16)        // data_size = 4 bytes
        | (1 << 20)        // pad_enable
        | (2 << 22);       // pad_interval: every 8 dwords (pad_amount=0 -> 1 dword)
  g1[1] = 0;               // atomic_barrier=0, tensor_dim0 low16 (dim0 = 1<<30)
  g1[2] = 0x4000;          // tensor_dim0 hi16, tensor_dim1 low16 (dim1 = 1<<30)
  g1[3] = 0x4000 | (TILE_C << 16);  // tensor_dim1 hi16, tile_dim0 = 8
  g1[4] = SS - 1;          // tile_dim1 = 511 rows, tile_dim2 = 0
  g1[5] = SS;              // tensor_dim0_stride = 512 (low32)
  g1[6] = 0;
  g1[7] = 0;
  i32x4 z4 = {0, 0, 0, 0};
#if __clang_major__ >= 23
  i32x8 z8 = {0, 0, 0, 0, 0, 0, 0, 0};
  __builtin_amdgcn_tensor_load_to_lds(g0, g1, z4, z4, z8, 0);
#else
  __builtin_amdgcn_tensor_load_to_lds(g0, g1, z4, z4, 0);
#endif
}
#endif  // USE_TDM

__device__ __forceinline__ void dp_step(int i, int j, float sh, float& myspan,
                                        float* red_val, int* red_idx,
                                        float* bc_val, int* span_trace_b) {
  bool valid = (j < i);                     // j<i implies j<=510, row j+1 in range
  float cand = valid ? (myspan + sh) : NEG_INF;
  int cidx = valid ? j : 0x7fffffff;        // losers carry huge index (first-max tiebreak)
#pragma unroll
  for (int off = 16; off > 0; off >>= 1) {  // wave32 shuffle max/argmax
    float ov = __shfl_xor(cand, off, 32);
    int oi = __shfl_xor(cidx, off, 32);
    if (ov > cand || (ov == cand && oi < cidx)) { cand = ov; cidx = oi; }
  }
  int wv = j >> 5, ln = j & 31;
  if (ln == 0) { red_val[wv] = cand; red_idx[wv] = cidx; }
  __syncthreads();
  if (wv == 0) {
    float v = (ln < 16) ? red_val[ln] : NEG_INF;
    int ix = (ln < 16) ? red_idx[ln] : 0x7fffffff;
#pragma unroll
    for (int off = 8; off > 0; off >>= 1) {
      float ov = __shfl_xor(v, off, 32);
      int oi = __shfl_xor(ix, off, 32);
      if (ov > v || (ov == v && oi < ix)) { v = ov; ix = oi; }
    }
    if (ln == 0) { *bc_val = v; span_trace_b[i] = ix; }
  }
  __syncthreads();
  if (j == i) myspan = *bc_val;             // thread i now owns span[i]
}

__global__ __launch_bounds__(512) void span_dp_kernel(
    const float* __restrict__ label_table,
    float* __restrict__ span_table,
    int* __restrict__ span_trace) {
  const int b = blockIdx.x;
  const int j = threadIdx.x;
  const float* lt = label_table + (size_t)b * SS * SS;

#if USE_TDM
  __shared__ float tile[2][TILE_DWORDS];
  // Issue chunk 0 DMA (rows 1..511, cols 0..7) as early as possible (wave 0).
  if (j < 32) tdm_issue(lt + SS, (unsigned)(size_t)&tile[0][0]);
#endif

  __shared__ float red_val[16];
  __shared__ int red_idx[16];
  __shared__ float bc_val;
  int* st_b = span_trace + b * SS;

  float myspan = (j == 0) ? lt[0] : NEG_INF;  // span0[:,0] = label_table[:,0,0]
  if (j == 0) st_b[0] = -1;                   // SOS_TRACE

#if USE_TDM
  for (int c = 0; c < NCHUNK; ++c) {
    if (j < 32) {
      if (c + 1 < NCHUNK) {
        tdm_issue(lt + SS + (c + 1) * TILE_C, (unsigned)(size_t)&tile[(c + 1) & 1][0]);
        __builtin_amdgcn_s_wait_tensorcnt(1);  // in-order: chunk c complete
      } else {
        __builtin_amdgcn_s_wait_tensorcnt(0);
      }
    }
    __syncthreads();  // publish tile[c&1] to all waves
    const float* buf = &tile[c & 1][0];
    const int i0 = c * TILE_C;
    for (int i = i0; i < i0 + TILE_C; ++i) {
      if (i == 0) continue;  // uniform skip; DP starts at i=1
      float sh = buf[j * TILE_STRIDE + (i - i0)];  // lt[b, j+1, i], conflict-free
      dp_step(i, j, sh, myspan, red_val, red_idx, &bc_val, st_b);
    }
  }
#else
  for (int i = 1; i < SS; ++i) {
    float sh = (j < SS - 1) ? lt[(size_t)(j + 1) * SS + i] : NEG_INF;
    if (j < SS - 1 && i + 16 < SS)
      __builtin_prefetch(&lt[(size_t)(j + 1) * SS + i + 16], 0, 0);
    dp_step(i, j, sh, myspan, red_val, red_idx, &bc_val, st_b);
  }
#endif
  span_table[b * SS + j] = myspan;  // thread 0 kept span0 value
}

// ---------------------------------------------------------------------------
// Kernel 1: max/argmax over last axis (K=32). One thread per row of 32 floats.
// 1 GiB streamed non-temporally; bandwidth bound (~46us floor @ 23.3 TB/s).
// ---------------------------------------------------------------------------
__global__ __launch_bounds__(256) void label_reduce_kernel(
    const float* __restrict__ score,
    float* __restrict__ label_table,
    int* __restrict__ label_trace) {
  size_t r = (size_t)blockIdx.x * 256 + threadIdx.x;  // row id < B*S*S
  const f32x4* p = (const f32x4*)(score + r * (size_t)KK);
  float best = -__builtin_inff();
  int bidx = 0;
#pragma unroll
  for (int q = 0; q < KK / 4; ++q) {
    f32x4 v = __builtin_nontemporal_load(&p[q]);
    if (v.x > best) { best = v.x; bidx = 4 * q + 0; }
    if (v.y > best) { best = v.y; bidx = 4 * q + 1; }
    if (v.z > best) { best = v.z; bidx = 4 * q + 2; }
    if (v.w > best) { best = v.w; bidx = 4 * q + 3; }
  }
  label_table[r] = best;                               // reread by kernel 2 (RT)
  __builtin_nontemporal_store(bidx, &label_trace[r]);  // write-once (NT)
}

// ---------------------------------------------------------------------------
extern "C" void kernel_launch(void* const* d_in, const int* in_sizes, int n_in,
                              void* d_out, int out_size, void* d_ws, size_t ws_size,
                              hipStream_t stream) {
  (void)in_sizes; (void)n_in; (void)out_size; (void)ws_size;
  const float* score = (const float*)d_in[0];
  float* out_f = (float*)d_out;
  // Tuple return order: span_table(f32 32x512), label_trace(i32 32x512x512),
  // span_trace(i32 32x512) — concatenated flat; int segments written as i32.
  float* span_table = out_f;
  int* label_trace = (int*)(out_f + (size_t)BB * SS);
  int* span_trace = (int*)(out_f + (size_t)BB * SS + (size_t)BB * SS * SS);
  float* label_table = (float*)d_ws;  // 32 MiB scratch

  size_t rows = (size_t)BB * SS * SS;  // 8,388,608
  label_reduce_kernel<<<dim3((unsigned)(rows / 256)), dim3(256), 0, stream>>>(
      score, label_table, label_trace);
  span_dp_kernel<<<dim3(BB), dim3(512), 0, stream>>>(
      label_table, span_table, span_trace);
}